// PointerNetwork_4406636445895
// MI455X (gfx1250) — compile-verified
//
#include <hip/hip_runtime.h>
#include <math.h>

#define N 4096
#define E 128
#define H 256
#define G4H 1024
#define TT 16   // decoder steps per attention block

typedef float v2f __attribute__((ext_vector_type(2)));
typedef float v8f __attribute__((ext_vector_type(8)));

__device__ __forceinline__ float fast_sigmoid(float x) {
    return 1.0f / (1.0f + __expf(-x));
}
__device__ __forceinline__ float fast_tanh(float x) {
#if __has_builtin(__builtin_amdgcn_tanhf)
    return __builtin_amdgcn_tanhf(x);
#else
    float e = __expf(2.0f * x);
    return 1.0f - 2.0f / (e + 1.0f);
#endif
}

// ---------------------------------------------------------------------------
// K0: embeddings (enc path + decoder teacher-forced input sequence) + bias sums
// ---------------------------------------------------------------------------
__global__ void embed_kernel(const float* __restrict__ x,            // (N,2)
                             const int*   __restrict__ tour,         // (N,)
                             const float* __restrict__ encW,         // (E,2)
                             const float* __restrict__ encb,         // (E,)
                             const float* __restrict__ cityW,        // (E,2)
                             const float* __restrict__ cityb,        // (E,)
                             const float* __restrict__ start_token,  // (E,)
                             const float* __restrict__ enc_bih,
                             const float* __restrict__ enc_bhh,
                             const float* __restrict__ dec_bih,
                             const float* __restrict__ dec_bhh,
                             float* __restrict__ emb,                // (N,E)
                             float* __restrict__ dec_in,             // (N,E)
                             float* __restrict__ bias_enc,           // (4H,)
                             float* __restrict__ bias_dec)           // (4H,)
{
    int g = blockIdx.x * blockDim.x + threadIdx.x;   // 0 .. N*E-1
    int i = g >> 7;        // row
    int e = g & 127;       // feature

    // encoder embedding
    emb[g] = x[i * 2] * encW[e * 2] + x[i * 2 + 1] * encW[e * 2 + 1] + encb[e];

    // decoder teacher-forced input: t==0 -> start_token, else city_embs[tour[t-1]]
    if (i == 0) {
        dec_in[g] = start_token[e];
    } else {
        int j = tour[i - 1];
        dec_in[g] = x[j * 2] * cityW[e * 2] + x[j * 2 + 1] * cityW[e * 2 + 1] + cityb[e];
    }

    if (g < G4H)               bias_enc[g]        = enc_bih[g] + enc_bhh[g];
    else if (g < 2 * G4H)      bias_dec[g - G4H]  = dec_bih[g - G4H] + dec_bhh[g - G4H];
}

// ---------------------------------------------------------------------------
// K1: f32 WMMA GEMM:  C(M x Nc) = A(M x K) @ W(Nc x K)^T  [+ bias]
// Optional transposed store: C[col*ldT + row] (for enc_projT).
// One 16x16 output tile per wave; 8 waves per block.
// ---------------------------------------------------------------------------
__global__ void gemm_wmma_f32(const float* __restrict__ A,
                              const float* __restrict__ W,
                              const float* __restrict__ bias0,
                              float* __restrict__ C,
                              int Ncols, int K, int transposeOut, int ldT)
{
    int lane = threadIdx.x & 31;
    int wave = threadIdx.x >> 5;
    int tile = blockIdx.x * 8 + wave;
    int tilesPerRow = Ncols >> 4;
    int tr = tile / tilesPerRow;
    int tc = tile - tr * tilesPerRow;
    int rowBase = tr << 4;
    int colBase = tc << 4;

    int m  = lane & 15;     // row (A) / col (B) handled by this lane
    int hi = lane >> 4;     // K-pair selector: 0 -> K {0,1}, 1 -> K {2,3}

    const float* aptr = A + (size_t)(rowBase + m) * K + 2 * hi;
    const float* bptr = W + (size_t)(colBase + m) * K + 2 * hi;

    v8f acc = {};
    for (int k0 = 0; k0 < K; k0 += 4) {
        float2 av = *(const float2*)(aptr + k0);
        float2 bv = *(const float2*)(bptr + k0);
        v2f a; a.x = av.x; a.y = av.y;
        v2f b; b.x = bv.x; b.y = bv.y;
        acc = __builtin_amdgcn_wmma_f32_16x16x4_f32(
            /*neg_a=*/false, a, /*neg_b=*/false, b,
            /*c_mod=*/(short)0, acc, /*reuse_a=*/false, /*reuse_b=*/false);
    }

    int col = colBase + m;                 // D layout: lane holds column (lane&15)
    float badd = bias0 ? bias0[col] : 0.0f;
#pragma unroll
    for (int r = 0; r < 8; ++r) {
        int row = rowBase + r + 8 * hi;    // VGPR r: M = r (lanes 0-15) / r+8 (lanes 16-31)
        float val = acc[r] + badd;
        if (transposeOut) C[(size_t)col * ldT + row] = val;
        else              C[(size_t)row * Ncols + col] = val;
    }
}

// ---------------------------------------------------------------------------
// K2/K3: sequential LSTM over T steps. One block of 1024 threads.
// xW already holds x@Wih.T + (bih+bhh) for every step; only h@Whh.T remains.
// 8 independent float4 FMA chains per gate (chain depth 32), 2 barriers/step.
// Gate order: i, f, g, o (PyTorch).
// ---------------------------------------------------------------------------
__global__ void lstm_seq_kernel(const float* __restrict__ xW,     // (T, 4H)
                                const float* __restrict__ Whh,    // (4H, H)
                                const float* __restrict__ hc0,    // (2H,) or null
                                float* __restrict__ h_out,        // (T, H)
                                float* __restrict__ hc_final,     // (2H,)
                                int T)
{
    __shared__ __align__(16) float h_s[H];
    __shared__ float pre[G4H];
    int tid = threadIdx.x;

    float c = 0.0f;
    if (tid < H) {
        h_s[tid] = hc0 ? hc0[tid] : 0.0f;
        c        = hc0 ? hc0[H + tid] : 0.0f;
    }
    __syncthreads();

    const float4* w4 = (const float4*)(Whh + (size_t)tid * H);   // 64 float4s
    const float4* h4 = (const float4*)h_s;

    for (int t = 0; t < T; ++t) {
        float s[8];
#pragma unroll
        for (int ch = 0; ch < 8; ++ch) s[ch] = 0.0f;
#pragma unroll
        for (int k = 0; k < 8; ++k) {
#pragma unroll
            for (int ch = 0; ch < 8; ++ch) {
                float4 w = w4[k + 8 * ch];
                float4 h = h4[k + 8 * ch];
                s[ch] = fmaf(w.x, h.x, s[ch]);
                s[ch] = fmaf(w.y, h.y, s[ch]);
                s[ch] = fmaf(w.z, h.z, s[ch]);
                s[ch] = fmaf(w.w, h.w, s[ch]);
            }
        }
        float acc = xW[(size_t)t * G4H + tid] +
                    (((s[0] + s[1]) + (s[2] + s[3])) + ((s[4] + s[5]) + (s[6] + s[7])));
        pre[tid] = acc;
        __syncthreads();   // barrier A: pre complete, all h_s reads done

        if (tid < H) {
            float ig = fast_sigmoid(pre[tid]);
            float fg = fast_sigmoid(pre[H + tid]);
            float gg = fast_tanh  (pre[2 * H + tid]);
            float og = fast_sigmoid(pre[3 * H + tid]);
            c  = fg * c + ig * gg;
            float hn = og * fast_tanh(c);
            h_out[(size_t)t * H + tid] = hn;
            h_s[tid] = hn;
        }
        __syncthreads();   // barrier B: h_s updated before next dot
    }
    if (tid < H) {
        hc_final[tid]     = h_s[tid];
        hc_final[H + tid] = c;
    }
}

// ---------------------------------------------------------------------------
// K6: attention scores e[t][j] = sum_k v[k]*tanh(projT[k][j] + q[t][k]),
// masked with -inf for j < t (tour == arange). Writes raw masked scores.
// Block: 256 threads -> 256 j's, TT t's. Grid: (N/256, N/TT).
// ---------------------------------------------------------------------------
__global__ void attn_scores_kernel(const float* __restrict__ projT,  // (H, N)
                                   const float* __restrict__ q,      // (N, H)
                                   const float* __restrict__ v,      // (H,)
                                   float* __restrict__ out)          // (N, N)
{
    __shared__ float q_s[TT * H];
    __shared__ float v_s[H];
    int tid = threadIdx.x;
    int tBase = blockIdx.y * TT;

    for (int idx = tid; idx < TT * H; idx += 256)
        q_s[idx] = q[(size_t)tBase * H + idx];
    if (tid < H) v_s[tid] = v[tid];
    __syncthreads();

    int j = blockIdx.x * 256 + tid;
    float acc[TT];
#pragma unroll
    for (int tt = 0; tt < TT; ++tt) acc[tt] = 0.0f;

    for (int k = 0; k < H; ++k) {
        float p  = projT[(size_t)k * N + j];   // coalesced
        float vk = v_s[k];
#pragma unroll
        for (int tt = 0; tt < TT; ++tt)
            acc[tt] = fmaf(vk, fast_tanh(p + q_s[tt * H + k]), acc[tt]);
    }
#pragma unroll
    for (int tt = 0; tt < TT; ++tt) {
        int t = tBase + tt;
        out[(size_t)t * N + j] = (j < t) ? -INFINITY : acc[tt];
    }
}

// ---------------------------------------------------------------------------
// K7: in-place masked log_softmax per row; row cached in LDS (16KB).
// ---------------------------------------------------------------------------
__global__ void row_log_softmax_kernel(float* __restrict__ out)
{
    __shared__ float rowe[N];
    __shared__ float red[256];
    int t = blockIdx.x;
    int tid = threadIdx.x;
    size_t base = (size_t)t * N;

    for (int idx = tid; idx < N; idx += 256) rowe[idx] = out[base + idx];
    __syncthreads();

    float m = -INFINITY;
    for (int idx = tid; idx < N; idx += 256) m = fmaxf(m, rowe[idx]);
    red[tid] = m; __syncthreads();
    for (int s = 128; s > 0; s >>= 1) {
        if (tid < s) red[tid] = fmaxf(red[tid], red[tid + s]);
        __syncthreads();
    }
    m = red[0];
    __syncthreads();

    float sum = 0.0f;
    for (int idx = tid; idx < N; idx += 256) sum += __expf(rowe[idx] - m);
    red[tid] = sum; __syncthreads();
    for (int s = 128; s > 0; s >>= 1) {
        if (tid < s) red[tid] += red[tid + s];
        __syncthreads();
    }
    float lse = m + __logf(red[0]);

    for (int idx = tid; idx < N; idx += 256) out[base + idx] = rowe[idx] - lse;
}

// ---------------------------------------------------------------------------
// K8: append tour (as float) after log_probs
// ---------------------------------------------------------------------------
__global__ void tour_copy_kernel(const int* __restrict__ tour, float* __restrict__ out_tail)
{
    int i = blockIdx.x * blockDim.x + threadIdx.x;
    if (i < N) out_tail[i] = (float)tour[i];
}

// ---------------------------------------------------------------------------
extern "C" void kernel_launch(void* const* d_in, const int* in_sizes, int n_in,
                              void* d_out, int out_size, void* d_ws, size_t ws_size,
                              hipStream_t stream) {
    const float* x           = (const float*)d_in[0];
    const int*   tour        = (const int*)  d_in[1];
    const float* enc_embed_W = (const float*)d_in[2];
    const float* enc_embed_b = (const float*)d_in[3];
    const float* enc_Wih     = (const float*)d_in[4];
    const float* enc_Whh     = (const float*)d_in[5];
    const float* enc_bih     = (const float*)d_in[6];
    const float* enc_bhh     = (const float*)d_in[7];
    const float* dec_Wih     = (const float*)d_in[8];
    const float* dec_Whh     = (const float*)d_in[9];
    const float* dec_bih     = (const float*)d_in[10];
    const float* dec_bhh     = (const float*)d_in[11];
    const float* att_W1      = (const float*)d_in[12];
    const float* att_W2      = (const float*)d_in[13];
    const float* att_v       = (const float*)d_in[14];
    const float* city_W      = (const float*)d_in[15];
    const float* city_b      = (const float*)d_in[16];
    const float* start_token = (const float*)d_in[17];

    float* out = (float*)d_out;
    float* ws  = (float*)d_ws;

    size_t o = 0;
    float* emb      = ws + o; o += (size_t)N * E;      // 4096x128
    float* dec_in   = ws + o; o += (size_t)N * E;      // 4096x128
    float* bias_enc = ws + o; o += G4H;
    float* bias_dec = ws + o; o += G4H;
    float* xW_enc   = ws + o; o += (size_t)N * G4H;    // 4096x1024
    float* xW_dec   = ws + o; o += (size_t)N * G4H;    // 4096x1024
    float* enc_out  = ws + o; o += (size_t)N * H;      // 4096x256
    float* hc       = ws + o; o += 2 * H;
    float* projT    = ws + o; o += (size_t)H * N;      // 256x4096 (transposed)
    float* hx_all   = ws + o; o += (size_t)N * H;      // 4096x256
    float* q_all    = ws + o; o += (size_t)N * H;      // 4096x256

    // K0: embeddings + bias sums
    embed_kernel<<<(N * E) / 256, 256, 0, stream>>>(
        x, tour, enc_embed_W, enc_embed_b, city_W, city_b, start_token,
        enc_bih, enc_bhh, dec_bih, dec_bhh, emb, dec_in, bias_enc, bias_dec);

    // K1: input projections (WMMA f32)  — 4096x1024, K=128
    gemm_wmma_f32<<<(N / 16) * (G4H / 16) / 8, 256, 0, stream>>>(
        emb, enc_Wih, bias_enc, xW_enc, G4H, E, 0, 0);
    gemm_wmma_f32<<<(N / 16) * (G4H / 16) / 8, 256, 0, stream>>>(
        dec_in, dec_Wih, bias_dec, xW_dec, G4H, E, 0, 0);

    // K2: encoder recurrence (sequential)
    lstm_seq_kernel<<<1, 1024, 0, stream>>>(xW_enc, enc_Whh, nullptr, enc_out, hc, N);

    // K4: enc_projT = (enc_out @ W1.T)^T  — stored k-major for coalesced attention
    gemm_wmma_f32<<<(N / 16) * (H / 16) / 8, 256, 0, stream>>>(
        enc_out, att_W1, nullptr, projT, H, H, 1, N);

    // K3: decoder recurrence, seeded with encoder final (h,c)
    lstm_seq_kernel<<<1, 1024, 0, stream>>>(xW_dec, dec_Whh, hc, hx_all, hc, N);

    // K5: q = hx @ W2.T
    gemm_wmma_f32<<<(N / 16) * (H / 16) / 8, 256, 0, stream>>>(
        hx_all, att_W2, nullptr, q_all, H, H, 0, 0);

    // K6: attention scores (masked) -> d_out
    attn_scores_kernel<<<dim3(N / 256, N / TT), 256, 0, stream>>>(projT, q_all, att_v, out);

    // K7: row-wise log_softmax in-place
    row_log_softmax_kernel<<<N, 256, 0, stream>>>(out);

    // K8: tour appended as float
    tour_copy_kernel<<<N / 256, 256, 0, stream>>>(tour, out + (size_t)N * N);
}